// CountMambaModel_24283745091709
// MI455X (gfx1250) — compile-verified
//
#include <hip/hip_runtime.h>
#include <hip/hip_bf16.h>
#include <math.h>

// ---------------- model constants ----------------
#define BSZ   32
#define PP    16
#define WW    6
#define DD    768
#define DEPTH 12
#define KK    5
#define DI    1536
#define DCONV 1792
#define NH    8
#define HD    192
#define DSZ   128
#define NC    100
#define ZXW   (2*DI + 2*DSZ + NH)   // 3336
#define KMAX  (DD*KK)               // 3840 (largest K, for zero-row)

typedef __attribute__((ext_vector_type(2))) float v2f;
typedef __attribute__((ext_vector_type(8))) float v8f;

// ============================================================
// fp32 WMMA GEMM, 16(M) x 32(N) per wave (two accumulators):
//   A  : M x K row-major (weights)
//   Bt : N x K row-major (activations)
//   Out: N x M row-major (ldc = M)
// M-tail rows are redirected to a pre-zeroed row (zrow) so the
// inner loop has no selects. A is loaded once per two WMMAs.
// ============================================================
__global__ __launch_bounds__(32)
void k_gemm_wmma_f32(const float* __restrict__ A,
                     const float* __restrict__ Bt,
                     float* __restrict__ Out,
                     int M, int N, int K,
                     const float* __restrict__ bias,
                     const float* __restrict__ scale,
                     const float* __restrict__ shift,
                     int relu, int accum,
                     const float* __restrict__ zrow)
{
    const int tm   = blockIdx.x;          // M tile (16 rows)
    const int tn   = blockIdx.y;          // N tile (32 cols)
    const int lane = threadIdx.x & 31;
    const int half = lane >> 4;           // 0: K pair {0,1}, 1: K pair {2,3}
    const int l15  = lane & 15;

    const int mA   = tm * 16 + l15;
    const bool mok = (mA < M);
    const float* __restrict__ arow  = mok ? (A + (size_t)mA * K) : zrow;
    const float* __restrict__ brow0 = Bt + (size_t)(tn * 32 + l15) * K;
    const float* __restrict__ brow1 = brow0 + (size_t)16 * K;

    v8f acc0 = {0.f,0.f,0.f,0.f,0.f,0.f,0.f,0.f};
    v8f acc1 = {0.f,0.f,0.f,0.f,0.f,0.f,0.f,0.f};

    int k0 = 0;
    for (; k0 + 32 <= K; k0 += 32) {
        if (k0 + 256 < K) __builtin_prefetch(arow + k0 + 256, 0, 0);
        #pragma unroll
        for (int kk = 0; kk < 32; kk += 4) {
            const int kb = k0 + kk + half * 2;
            v2f a, b0, b1;
            a.x  = arow[kb];   a.y  = arow[kb + 1];
            b0.x = brow0[kb];  b0.y = brow0[kb + 1];
            b1.x = brow1[kb];  b1.y = brow1[kb + 1];
            acc0 = __builtin_amdgcn_wmma_f32_16x16x4_f32(
                       false, a, false, b0, (short)0, acc0, false, false);
            acc1 = __builtin_amdgcn_wmma_f32_16x16x4_f32(
                       false, a, false, b1, (short)0, acc1, false, false);
        }
    }
    for (; k0 < K; k0 += 4) {            // remainder (K=16 patch GEMM)
        const int kb = k0 + half * 2;
        v2f a, b0, b1;
        a.x  = arow[kb];   a.y  = arow[kb + 1];
        b0.x = brow0[kb];  b0.y = brow0[kb + 1];
        b1.x = brow1[kb];  b1.y = brow1[kb + 1];
        acc0 = __builtin_amdgcn_wmma_f32_16x16x4_f32(
                   false, a, false, b0, (short)0, acc0, false, false);
        acc1 = __builtin_amdgcn_wmma_f32_16x16x4_f32(
                   false, a, false, b1, (short)0, acc1, false, false);
    }

    #pragma unroll
    for (int r = 0; r < 8; ++r) {
        const int m = tm * 16 + r + half * 8;
        if (m < M) {
            float v0 = acc0[r];
            float v1 = acc1[r];
            if (bias)  { float bv = bias[m]; v0 += bv; v1 += bv; }
            if (scale) { float sc = scale[m], sh = shift[m];
                         v0 = v0 * sc + sh; v1 = v1 * sc + sh; }
            if (relu)  { v0 = fmaxf(v0, 0.f); v1 = fmaxf(v1, 0.f); }
            const int n0 = tn * 32 + l15;
            const size_t o0 = (size_t)n0 * (size_t)M + (size_t)m;
            const size_t o1 = o0 + (size_t)16 * (size_t)M;
            if (accum) { Out[o0] += v0; Out[o1] += v1; }
            else       { Out[o0]  = v0; Out[o1]  = v1; }
        }
    }
}

// ---------------- small helpers ----------------
__device__ __forceinline__ float silu_f(float v) { return v / (1.f + expf(-v)); }

__global__ void k_zerof(float* __restrict__ p, int n)
{
    int i = blockIdx.x * blockDim.x + threadIdx.x;
    if (i < n) p[i] = 0.f;
}

// x (B,1,P,W) -> xT (B*W, P)
__global__ void k_transpose_x(const float* __restrict__ x, float* __restrict__ xT)
{
    int idx = blockIdx.x * blockDim.x + threadIdx.x;
    if (idx >= BSZ * WW * PP) return;
    int p = idx % PP;
    int n = idx / PP;            // b*W + w
    int b = n / WW, w = n % WW;
    xT[idx] = x[(size_t)b * (PP * WW) + p * WW + w];
}

// BN fold: scale = g*rsqrt(rv+eps); shift = (conv_b - rm)*scale + bn_b
__global__ void k_bncoef(const float* __restrict__ cb, const float* __restrict__ g,
                         const float* __restrict__ bb, const float* __restrict__ rm,
                         const float* __restrict__ rv,
                         float* __restrict__ S, float* __restrict__ T)
{
    int m = blockIdx.x * blockDim.x + threadIdx.x;
    if (m >= DD) return;
    float s = g[m] * rsqrtf(rv[m] + 1e-5f);
    S[m] = s;
    T[m] = (cb[m] - rm[m]) * s + bb[m];
}

// im2col: col[(b*L+l), din*5+kk] with width = [state(4) | h(L)]
__global__ void k_im2col(const float* __restrict__ h,     // (B*L, D) n-major
                         const float* __restrict__ state, // (B, D, 4)
                         float* __restrict__ col, int L)
{
    int idx = blockIdx.x * blockDim.x + threadIdx.x;
    int total = BSZ * L * DD * KK;
    if (idx >= total) return;
    int kk  = idx % KK;
    int t   = idx / KK;
    int din = t % DD;
    int n   = t / DD;                 // b*L + l
    int b = n / L, l = n % L;
    int w = l + kk;                   // 0..L+3
    float v;
    if (w < 4) v = state[((size_t)b * DD + din) * 4 + w];
    else       v = h[(size_t)(b * L + (w - 4)) * DD + din];
    col[(size_t)n * (DD * KK) + (size_t)din * KK + kk] = v;
}

// maxpool width 3: (B*6, D) -> (B*2, D)
__global__ void k_pool3(const float* __restrict__ h, float* __restrict__ o)
{
    int idx = blockIdx.x * blockDim.x + threadIdx.x;
    if (idx >= BSZ * 2 * DD) return;
    int d = idx % DD;
    int n = idx / DD;                 // b*2 + j
    int b = n / 2, j = n % 2;
    const float* p = h + (size_t)(b * 6 + j * 3) * DD + d;
    o[idx] = fmaxf(fmaxf(p[0], p[DD]), p[2 * DD]);
}

// maxpool width 2 + positional embedding -> tok (B, D)
__global__ void k_pool2_pos(const float* __restrict__ h, const float* __restrict__ pe,
                            const int* __restrict__ pos, float* __restrict__ tok)
{
    int idx = blockIdx.x * blockDim.x + threadIdx.x;
    if (idx >= BSZ * DD) return;
    int d = idx % DD;
    int b = idx / DD;
    float v = fmaxf(h[(size_t)(b * 2) * DD + d], h[(size_t)(b * 2 + 1) * DD + d]);
    int clip = pos[0] < 299 ? pos[0] : 299;
    tok[idx] = v + pe[(size_t)(1 + clip) * DD + d];
}

// mamba depthwise conv step + silu: xc (B, DCONV)
__global__ void k_conv_step(const float* __restrict__ zx, const float* __restrict__ mcs,
                            const float* __restrict__ cw, const float* __restrict__ cb,
                            float* __restrict__ xc)
{
    int idx = blockIdx.x * blockDim.x + threadIdx.x;
    if (idx >= BSZ * DCONV) return;
    int c = idx % DCONV, b = idx / DCONV;
    const float* st = mcs + ((size_t)b * DCONV + c) * 4;
    const float* w  = cw + (size_t)c * 4;
    float x = zx[(size_t)b * ZXW + DI + c];
    float v = st[1] * w[0] + st[2] * w[1] + st[3] * w[2] + x * w[3] + cb[c];
    xc[idx] = silu_f(v);
}

// per (b): bcdot = dot(B,C); per (b,h): dt = softplus(dt+dtb), dA = exp(dt*-exp(A_log))
__global__ void k_bh_prep(const float* __restrict__ zx, const float* __restrict__ xc,
                          const float* __restrict__ A_log, const float* __restrict__ dtb,
                          float* __restrict__ dtA, float* __restrict__ bcdot)
{
    __shared__ float red[128];
    int b = blockIdx.x, t = threadIdx.x;   // 128 threads
    float Bv = xc[(size_t)b * DCONV + DI + t];
    float Cv = xc[(size_t)b * DCONV + DI + DSZ + t];
    red[t] = Bv * Cv;
    __syncthreads();
    for (int s = 64; s > 0; s >>= 1) { if (t < s) red[t] += red[t + s]; __syncthreads(); }
    if (t == 0) bcdot[b] = red[0];
    if (t < NH) {
        float dtr = zx[(size_t)b * ZXW + 2*DI + 2*DSZ + t] + dtb[t];
        float dt  = (dtr > 20.f) ? dtr : log1pf(expf(dtr));
        float dA  = expf(dt * -expf(A_log[t]));
        dtA[b * NH + t]             = dt;
        dtA[BSZ * NH + b * NH + t]  = dA;
    }
}

// SSM: y[b,h,p] = dA*dot(ssm[b,h,p,:],C[b,:]) + x*(dt*bcdot + Dp[h]); one wave per (b,h,p)
__global__ void k_ssm(const float* __restrict__ ssm,   // (B,NH,HD,DSZ)
                      const float* __restrict__ xc,    // (B,DCONV)
                      const float* __restrict__ dtA,   // dt[B*NH], dA at +B*NH
                      const float* __restrict__ bcdot, // (B)
                      const float* __restrict__ Dp,    // (NH)
                      float* __restrict__ y)           // (B,DI)
{
    int wave = (blockIdx.x * blockDim.x + threadIdx.x) >> 5;
    int lane = threadIdx.x & 31;
    if (wave >= BSZ * NH * HD) return;
    int p  = wave % HD;
    int bh = wave / HD;
    int h = bh % NH, b = bh / NH;

    const float4* s4 = reinterpret_cast<const float4*>(ssm + (size_t)wave * DSZ);
    const float4* c4 = reinterpret_cast<const float4*>(xc + (size_t)b * DCONV + DI + DSZ);
    float4 s = s4[lane];
    float4 c = c4[lane];
    float part = s.x * c.x + s.y * c.y + s.z * c.z + s.w * c.w;
    part += __shfl_xor(part, 16);
    part += __shfl_xor(part, 8);
    part += __shfl_xor(part, 4);
    part += __shfl_xor(part, 2);
    part += __shfl_xor(part, 1);
    if (lane == 0) {
        float dt = dtA[b * NH + h];
        float dA = dtA[BSZ * NH + b * NH + h];
        float x  = xc[(size_t)b * DCONV + h * HD + p];
        y[(size_t)b * DI + h * HD + p] = dA * part + x * (dt * bcdot[b] + Dp[h]);
    }
}

// gated RMSNorm: yn = (y*silu(z)) * rsqrt(mean(v^2)+eps) * nw
__global__ void k_gate_norm(const float* __restrict__ zx, const float* __restrict__ y,
                            const float* __restrict__ nw, float* __restrict__ yn)
{
    __shared__ float red[256];
    int b = blockIdx.x, t = threadIdx.x;   // 256 threads, 6 elems each
    float vals[6]; float ss = 0.f;
    #pragma unroll
    for (int j0 = 0; j0 < 6; ++j0) {
        int j = t + j0 * 256;
        float z = zx[(size_t)b * ZXW + j];
        float v = y[(size_t)b * DI + j] * silu_f(z);
        vals[j0] = v; ss += v * v;
    }
    red[t] = ss; __syncthreads();
    for (int s = 128; s > 0; s >>= 1) { if (t < s) red[t] += red[t + s]; __syncthreads(); }
    float scale = rsqrtf(red[0] / (float)DI + 1e-5f);
    #pragma unroll
    for (int j0 = 0; j0 < 6; ++j0) {
        int j = t + j0 * 256;
        yn[(size_t)b * DI + j] = vals[j0] * scale * nw[j];
    }
}

// final layernorm + history blend; writes history (B,D) into d_out+NC*B
__global__ void k_final(const float* __restrict__ tok, const float* __restrict__ hist_in,
                        const float* __restrict__ g, const float* __restrict__ bb,
                        const int* __restrict__ pos, float* __restrict__ hist_out)
{
    __shared__ float red[256];
    int b = blockIdx.x, t = threadIdx.x;   // 256 threads, 3 elems each
    float v[3]; float s = 0.f;
    #pragma unroll
    for (int j0 = 0; j0 < 3; ++j0) { v[j0] = tok[(size_t)b * DD + t + j0 * 256]; s += v[j0]; }
    red[t] = s; __syncthreads();
    for (int st = 128; st > 0; st >>= 1) { if (t < st) red[t] += red[t + st]; __syncthreads(); }
    float mu = red[0] / (float)DD;
    __syncthreads();
    float sv = 0.f;
    #pragma unroll
    for (int j0 = 0; j0 < 3; ++j0) { float d = v[j0] - mu; sv += d * d; }
    red[t] = sv; __syncthreads();
    for (int st = 128; st > 0; st >>= 1) { if (t < st) red[t] += red[t + st]; __syncthreads(); }
    float inv = rsqrtf(red[0] / (float)DD + 1e-5f);
    float pf  = (float)pos[0];
    #pragma unroll
    for (int j0 = 0; j0 < 3; ++j0) {
        int j = t + j0 * 256;
        float tt = (v[j0] - mu) * inv * g[j] + bb[j];
        hist_out[(size_t)b * DD + j] = (hist_in[(size_t)b * DD + j] * pf + tt) / (pf + 1.f);
    }
}

// ============================================================
extern "C" void kernel_launch(void* const* d_in, const int* in_sizes, int n_in,
                              void* d_out, int out_size, void* d_ws, size_t ws_size,
                              hipStream_t stream)
{
    (void)in_sizes; (void)n_in; (void)out_size; (void)ws_size;
    const float* x        = (const float*)d_in[0];
    const float* hist     = (const float*)d_in[1];
    const float* patch_w  = (const float*)d_in[2];
    const float* patch_b  = (const float*)d_in[3];
    const float* cnn_w    = (const float*)d_in[4];
    const float* cnn_b    = (const float*)d_in[5];
    const float* bn_g     = (const float*)d_in[6];
    const float* bn_b     = (const float*)d_in[7];
    const float* bn_rm    = (const float*)d_in[8];
    const float* bn_rv    = (const float*)d_in[9];
    const float* cnn_st   = (const float*)d_in[10];
    const float* pos_emb  = (const float*)d_in[11];
    const float* inproj_w = (const float*)d_in[12];
    const float* c1d_w    = (const float*)d_in[13];
    const float* c1d_b    = (const float*)d_in[14];
    const float* A_log    = (const float*)d_in[15];
    const float* dt_bias  = (const float*)d_in[16];
    const float* D_param  = (const float*)d_in[17];
    const float* norm_w   = (const float*)d_in[18];
    const float* outp_w   = (const float*)d_in[19];
    const float* mcs      = (const float*)d_in[20];
    const float* ssm      = (const float*)d_in[21];
    const float* fcn_g    = (const float*)d_in[22];
    const float* fcn_b    = (const float*)d_in[23];
    const float* fc_w     = (const float*)d_in[24];
    const float* fc_b     = (const float*)d_in[25];
    const int*   pos      = (const int*)d_in[26];
    float* out = (float*)d_out;

    // workspace layout (floats)
    float* ws = (float*)d_ws;
    size_t o = 0;
    float* xT   = ws + o; o += (size_t)BSZ * WW * PP;       // 3072
    float* bufA = ws + o; o += (size_t)BSZ * WW * DD;       // 147456
    float* bufB = ws + o; o += (size_t)BSZ * WW * DD;       // 147456
    float* col  = ws + o; o += (size_t)BSZ * WW * DD * KK;  // 737280
    float* pool = ws + o; o += (size_t)BSZ * 2 * DD;        // 49152
    float* tok  = ws + o; o += (size_t)BSZ * DD;            // 24576
    float* zx   = ws + o; o += (size_t)BSZ * ZXW;           // 106752
    float* xc   = ws + o; o += (size_t)BSZ * DCONV;         // 57344
    float* dtA  = ws + o; o += (size_t)2 * BSZ * NH;        // 512
    float* bcd  = ws + o; o += (size_t)BSZ;                 // 32
    float* ybuf = ws + o; o += (size_t)BSZ * DI;            // 49152
    float* ynb  = ws + o; o += (size_t)BSZ * DI;            // 49152
    float* bnS  = ws + o; o += DD;
    float* bnT  = ws + o; o += DD;
    float* zrow = ws + o; o += KMAX;                        // 3840 zero row

    float* logits   = out;                 // (B, NC)
    float* hist_out = out + BSZ * NC;      // (B, DD)

    // zero the OOB-row buffer (deterministic, every call)
    k_zerof<<<(KMAX + 255)/256, 256, 0, stream>>>(zrow, KMAX);

    // ---- patch embedding: h0 = x^T @ patch_w^T + b  -> (B*W, D)
    k_transpose_x<<<(BSZ*WW*PP + 255)/256, 256, 0, stream>>>(x, xT);
    {
        dim3 g(DD/16, (BSZ*WW)/32);
        k_gemm_wmma_f32<<<g, 32, 0, stream>>>(patch_w, xT, bufA,
            DD, BSZ*WW, PP, patch_b, nullptr, nullptr, 0, 0, zrow);
    }

    // ---- conv layers 0,1 at L=6
    const float* hin = bufA; float* hout = bufB;
    for (int i = 0; i < 2; ++i) {
        k_bncoef<<<(DD+255)/256, 256, 0, stream>>>(
            cnn_b + (size_t)i*DD, bn_g + (size_t)i*DD, bn_b + (size_t)i*DD,
            bn_rm + (size_t)i*DD, bn_rv + (size_t)i*DD, bnS, bnT);
        int tot = BSZ*6*DD*KK;
        k_im2col<<<(tot+255)/256, 256, 0, stream>>>(
            hin, cnn_st + (size_t)i*BSZ*DD*4, col, 6);
        dim3 g(DD/16, (BSZ*6)/32);
        k_gemm_wmma_f32<<<g, 32, 0, stream>>>(cnn_w + (size_t)i*DD*DD*KK, col, hout,
            DD, BSZ*6, DD*KK, nullptr, bnS, bnT, 1, 0, zrow);
        const float* tswap = hin; hin = hout; hout = (float*)tswap;
    }
    // hin now holds layer-1 output (bufA)
    k_pool3<<<(BSZ*2*DD + 255)/256, 256, 0, stream>>>(hin, pool);

    // ---- conv layers 2,3 at L=2
    hin = pool; hout = bufB;
    for (int i = 2; i < 4; ++i) {
        k_bncoef<<<(DD+255)/256, 256, 0, stream>>>(
            cnn_b + (size_t)i*DD, bn_g + (size_t)i*DD, bn_b + (size_t)i*DD,
            bn_rm + (size_t)i*DD, bn_rv + (size_t)i*DD, bnS, bnT);
        int tot = BSZ*2*DD*KK;
        k_im2col<<<(tot+255)/256, 256, 0, stream>>>(
            hin, cnn_st + (size_t)i*BSZ*DD*4, col, 2);
        dim3 g(DD/16, (BSZ*2)/32);
        k_gemm_wmma_f32<<<g, 32, 0, stream>>>(cnn_w + (size_t)i*DD*DD*KK, col, hout,
            DD, BSZ*2, DD*KK, nullptr, bnS, bnT, 1, 0, zrow);
        hin = hout; hout = (hout == bufB) ? bufA : bufB;
    }
    // hin = layer-3 output
    k_pool2_pos<<<(BSZ*DD + 255)/256, 256, 0, stream>>>(hin, pos_emb, pos, tok);

    // ---- 12 Mamba step layers
    for (int i = 0; i < DEPTH; ++i) {
        dim3 g1((ZXW + 15)/16, BSZ/32);
        k_gemm_wmma_f32<<<g1, 32, 0, stream>>>(inproj_w + (size_t)i*ZXW*DD, tok, zx,
            ZXW, BSZ, DD, nullptr, nullptr, nullptr, 0, 0, zrow);

        k_conv_step<<<(BSZ*DCONV + 255)/256, 256, 0, stream>>>(
            zx, mcs + (size_t)i*BSZ*DCONV*4,
            c1d_w + (size_t)i*DCONV*4, c1d_b + (size_t)i*DCONV, xc);

        k_bh_prep<<<BSZ, 128, 0, stream>>>(
            zx, xc, A_log + (size_t)i*NH, dt_bias + (size_t)i*NH, dtA, bcd);

        {
            int waves = BSZ * NH * HD;                 // 49152
            int blocks = (waves * 32 + 255) / 256;
            k_ssm<<<blocks, 256, 0, stream>>>(
                ssm + (size_t)i*BSZ*NH*HD*DSZ, xc, dtA, bcd,
                D_param + (size_t)i*NH, ybuf);
        }

        k_gate_norm<<<BSZ, 256, 0, stream>>>(zx, ybuf, norm_w + (size_t)i*DI, ynb);

        dim3 g2(DD/16, BSZ/32);
        k_gemm_wmma_f32<<<g2, 32, 0, stream>>>(outp_w + (size_t)i*DD*DI, ynb, tok,
            DD, BSZ, DI, nullptr, nullptr, nullptr, 0, /*accum=*/1, zrow);
    }

    // ---- head: layernorm + history blend, then fc
    k_final<<<BSZ, 256, 0, stream>>>(tok, hist, fcn_g, fcn_b, pos, hist_out);
    {
        dim3 g((NC + 15)/16, BSZ/32);
        k_gemm_wmma_f32<<<g, 32, 0, stream>>>(fc_w, hist_out, logits,
            NC, BSZ, DD, fc_b, nullptr, nullptr, 0, 0, zrow);
    }
}